// Memory_74852690035338
// MI455X (gfx1250) — compile-verified
//
#include <hip/hip_runtime.h>

// NTM memory addressing + read/write for B=64, N=8192, M=128 (fp32).
// Bandwidth-bound (~768 MB HBM traffic -> ~33us @ 23.3 TB/s). WMMA f32 16x16x4
// used for the content-addressing matvec (fp32-exact, free on the mem-bound pass).

static constexpr int Bc = 64;
static constexpr int Nc = 8192;
static constexpr int Mc = 128;
#define EPSV     1e-16f
#define EPS_COSV 1e-8f

typedef __attribute__((ext_vector_type(2))) float v2f;
typedef __attribute__((ext_vector_type(8))) float v8f;

#ifndef __has_builtin
#define __has_builtin(x) 0
#endif
#if __has_builtin(__builtin_amdgcn_wmma_f32_16x16x4_f32)
#define USE_WMMA_F32X4 1
#else
#define USE_WMMA_F32X4 0
#endif

__device__ __forceinline__ float waveRedSum(float v) {
#pragma unroll
  for (int o = 16; o > 0; o >>= 1) v += __shfl_xor(v, o, 32);
  return v;
}

// ---------- k0: nk[b] = max(||k[b]+eps||, 1e-8) ----------
__global__ void k0_knorm(const float* __restrict__ k, float* __restrict__ nk) {
  int b = blockIdx.x, lane = threadIdx.x;
  float s = 0.f;
#pragma unroll
  for (int i = 0; i < Mc / 32; ++i) {
    float v = k[b * Mc + lane + i * 32] + EPSV;
    s += v * v;
  }
  s = waveRedSum(s);
  if (lane == 0) nk[b] = fmaxf(sqrtf(s), EPS_COSV);
}

// ---------- k1: p[b,n] = exp(beta * cos(mem[b,n]+eps, k[b]+eps)), + block sums ----------
// Block = 256 threads = 8 waves; each wave owns a 16-row tile, K swept in 4-chunks.
__global__ void k1_content(const float* __restrict__ mem, const float* __restrict__ k,
                           const float* __restrict__ beta, const float* __restrict__ nk,
                           float* __restrict__ p, float* __restrict__ ps1) {
  const int tilesPerBatch = Nc / 128;  // 64
  int b = blockIdx.x / tilesPerBatch;
  int tile = blockIdx.x % tilesPerBatch;
  int t = threadIdx.x;
  int lane = t & 31, wave = t >> 5;

  __shared__ float lk[Mc];
  __shared__ float wsum[8];
  if (t < Mc) lk[t] = k[b * Mc + t] + EPSV;
  __syncthreads();

  int n0 = tile * 128 + wave * 16;
  const float* __restrict__ row =
      mem + ((size_t)b * Nc + (size_t)(n0 + (lane & 15))) * (size_t)Mc;
  // A-matrix f32 16x4 layout: lanes 0-15 hold K={0,1}, lanes 16-31 hold K={2,3}
  int kk = (lane >> 4) << 1;

  float nsq = 0.f;
#if USE_WMMA_F32X4
  v8f acc = {0.f, 0.f, 0.f, 0.f, 0.f, 0.f, 0.f, 0.f};
#else
  float dotp = 0.f;
#endif

#pragma unroll 8
  for (int kc = 0; kc < Mc / 4; ++kc) {
    int base = kc * 4;
    float a0 = row[base + kk] + EPSV;
    float a1 = row[base + kk + 1] + EPSV;
    float b0 = lk[base + kk];
    float b1 = lk[base + kk + 1];
    nsq += a0 * a0 + a1 * a1;
#if USE_WMMA_F32X4
    v2f A;  A.x = a0; A.y = a1;
    v2f Bv; Bv.x = b0; Bv.y = b1;   // k replicated across all 16 N-columns
    acc = __builtin_amdgcn_wmma_f32_16x16x4_f32(false, A, false, Bv, (short)0,
                                                acc, false, false);
#else
    dotp += a0 * b0 + a1 * b1;
#endif
  }

  // full row sumsq: lane L and L+16 hold halves of row (L&15)
  float nsqf = nsq + __shfl_xor(nsq, 16, 32);

#if USE_WMMA_F32X4
  // D layout: VGPR j, lanes 0-15 -> M=j; lanes 16-31 -> M=j+8 (all N cols equal).
  int j = lane & 7;
  float d = acc[0];
  if (j == 1) d = acc[1];
  if (j == 2) d = acc[2];
  if (j == 3) d = acc[3];
  if (j == 4) d = acc[4];
  if (j == 5) d = acc[5];
  if (j == 6) d = acc[6];
  if (j == 7) d = acc[7];
  bool writer = (lane < 8) || (lane >= 16 && lane < 24);
  int rowM = (lane < 16) ? lane : (lane - 8);  // writer lanes cover rows 0..15
#else
  float d = dotp + __shfl_xor(dotp, 16, 32);
  bool writer = (lane < 16);
  int rowM = lane & 15;
#endif
  float nsqw = __shfl(nsqf, rowM & 15, 32);  // row r's sumsq lives at lane r

  float pv = 0.f;
  if (writer) {
    float nm = fmaxf(sqrtf(nsqw), EPS_COSV);
    float cosv = d / (nm * nk[b]);
    pv = __expf(beta[b] * cosv);  // |beta*cos| <= ~5: safe without max-shift
    p[(size_t)b * Nc + (size_t)(n0 + rowM)] = pv;
  }
  float ws = waveRedSum(pv);
  if (lane == 0) wsum[wave] = ws;
  __syncthreads();
  if (t == 0) {
    float s = 0.f;
#pragma unroll
    for (int i = 0; i < 8; ++i) s += wsum[i];
    ps1[b * tilesPerBatch + tile] = s;
  }
}

// ---------- k2: S1[b] = sum of 64 partials ----------
__global__ void k2_red1(const float* __restrict__ ps1, float* __restrict__ S1) {
  int b = blockIdx.x, lane = threadIdx.x;
  float s = ps1[b * 64 + lane] + ps1[b * 64 + lane + 32];
  s = waveRedSum(s);
  if (lane == 0) S1[b] = s;
}

// ---------- k3: interpolate + circular shift + pow, with block sums ----------
__global__ void k3_shift(const float* __restrict__ p, const float* __restrict__ S1,
                         const float* __restrict__ wprev, const float* __restrict__ g,
                         const float* __restrict__ s3, const float* __restrict__ gamma,
                         float* __restrict__ wp, float* __restrict__ ps2) {
  const int tilesPerBatch = Nc / 256;  // 32
  int b = blockIdx.x / tilesPerBatch;
  int tile = blockIdx.x % tilesPerBatch;
  int t = threadIdx.x;
  int jidx = tile * 256 + t;

  float invS1 = 1.f / S1[b];
  float gb = g[b];
  float s0 = s3[b * 3 + 0], s1 = s3[b * 3 + 1], s2 = s3[b * 3 + 2];
  const float* __restrict__ pb = p + (size_t)b * Nc;
  const float* __restrict__ wb = wprev + (size_t)b * Nc;

  int jm = (jidx == 0) ? (Nc - 1) : (jidx - 1);
  int jp = (jidx == Nc - 1) ? 0 : (jidx + 1);
  float wgm = gb * (pb[jm] * invS1) + (1.f - gb) * wb[jm];
  float wg0 = gb * (pb[jidx] * invS1) + (1.f - gb) * wb[jidx];
  float wgp = gb * (pb[jp] * invS1) + (1.f - gb) * wb[jp];
  float what = s0 * wgm + s1 * wg0 + s2 * wgp;  // w_hat > 0 strictly
  float wpow = __powf(what, gamma[b]);
  wp[(size_t)b * Nc + jidx] = wpow;

  __shared__ float wsum[8];
  float ws = waveRedSum(wpow);
  int lane = t & 31, wave = t >> 5;
  if (lane == 0) wsum[wave] = ws;
  __syncthreads();
  if (t == 0) {
    float s = 0.f;
#pragma unroll
    for (int i = 0; i < 8; ++i) s += wsum[i];
    ps2[b * tilesPerBatch + tile] = s;
  }
}

// ---------- k4: S2[b] = sum of 32 partials ----------
__global__ void k4_red2(const float* __restrict__ ps2, float* __restrict__ S2) {
  int b = blockIdx.x, lane = threadIdx.x;
  float s = ps2[b * 32 + lane];
  s = waveRedSum(s);
  if (lane == 0) S2[b] = s;
}

// ---------- k5: fused read + erase/add (single streaming pass over memory) ----------
__global__ void k5_rw(const float* __restrict__ mem, const float* __restrict__ wp,
                      const float* __restrict__ S2, const float* __restrict__ e,
                      const float* __restrict__ a, float* __restrict__ newmem,
                      float* __restrict__ rp) {
  const int tilesPerBatch = Nc / 256;  // 32
  int b = blockIdx.x / tilesPerBatch;
  int tile = blockIdx.x % tilesPerBatch;
  int t = threadIdx.x;
  int c4 = t & 31;  // float4 column
  int rg = t >> 5;  // row group 0..7

  float inv = 1.f / (S2[b] + EPSV);
  const float4* __restrict__ e4p = reinterpret_cast<const float4*>(e + (size_t)b * Mc);
  const float4* __restrict__ a4p = reinterpret_cast<const float4*>(a + (size_t)b * Mc);
  float4 e4 = e4p[c4], a4 = a4p[c4];
  float4 racc = {0.f, 0.f, 0.f, 0.f};
  const float* __restrict__ wb = wp + (size_t)b * Nc;
  const float4* __restrict__ m4p = reinterpret_cast<const float4*>(mem);
  float4* __restrict__ o4p = reinterpret_cast<float4*>(newmem);
  int r0 = tile * 256;

#pragma unroll 4
  for (int it = 0; it < 32; ++it) {
    int rowIdx = r0 + it * 8 + rg;
    float w = wb[rowIdx] * inv;
    size_t off = ((size_t)b * Nc + (size_t)rowIdx) * (size_t)(Mc / 4) + c4;
    float4 m4 = m4p[off];
    racc.x += w * m4.x; racc.y += w * m4.y;
    racc.z += w * m4.z; racc.w += w * m4.w;
    float4 o4;
    o4.x = m4.x * (1.f - w * e4.x) + w * a4.x;
    o4.y = m4.y * (1.f - w * e4.y) + w * a4.y;
    o4.z = m4.z * (1.f - w * e4.z) + w * a4.z;
    o4.w = m4.w * (1.f - w * e4.w) + w * a4.w;
    o4p[off] = o4;
  }

  __shared__ float lr[8 * Mc];
  lr[rg * Mc + c4 * 4 + 0] = racc.x;
  lr[rg * Mc + c4 * 4 + 1] = racc.y;
  lr[rg * Mc + c4 * 4 + 2] = racc.z;
  lr[rg * Mc + c4 * 4 + 3] = racc.w;
  __syncthreads();
  if (t < Mc) {
    float s = 0.f;
#pragma unroll
    for (int i = 0; i < 8; ++i) s += lr[i * Mc + t];
    rp[((size_t)b * tilesPerBatch + tile) * Mc + t] = s;
  }
}

// ---------- k6: r[b,m] = sum over 32 block partials ----------
__global__ void k6_redr(const float* __restrict__ rp, float* __restrict__ r) {
  int b = blockIdx.x, m = threadIdx.x;
  float s = 0.f;
#pragma unroll 8
  for (int blk = 0; blk < 32; ++blk) s += rp[((size_t)b * 32 + blk) * Mc + m];
  r[b * Mc + m] = s;
}

extern "C" void kernel_launch(void* const* d_in, const int* in_sizes, int n_in,
                              void* d_out, int out_size, void* d_ws, size_t ws_size,
                              hipStream_t stream) {
  (void)in_sizes; (void)n_in; (void)out_size; (void)ws_size;
  const float* memory = (const float*)d_in[0];
  const float* k      = (const float*)d_in[1];
  const float* beta   = (const float*)d_in[2];
  const float* g      = (const float*)d_in[3];
  const float* s      = (const float*)d_in[4];
  const float* gamma  = (const float*)d_in[5];
  const float* w_prev = (const float*)d_in[6];
  const float* e      = (const float*)d_in[7];
  const float* a      = (const float*)d_in[8];

  float* out_r   = (float*)d_out;            // [B, M]
  float* out_mem = out_r + (size_t)Bc * Mc;  // [B, N, M]

  float* w  = (float*)d_ws;
  float* nk  = w; w += Bc;
  float* S1  = w; w += Bc;
  float* S2  = w; w += Bc;
  float* ps1 = w; w += (size_t)Bc * (Nc / 128);
  float* ps2 = w; w += (size_t)Bc * (Nc / 256);
  float* p   = w; w += (size_t)Bc * Nc;
  float* wp  = w; w += (size_t)Bc * Nc;
  float* rp  = w; w += (size_t)Bc * (Nc / 256) * Mc;  // ~5.3 MB total

  k0_knorm<<<Bc, 32, 0, stream>>>(k, nk);
  k1_content<<<Bc * (Nc / 128), 256, 0, stream>>>(memory, k, beta, nk, p, ps1);
  k2_red1<<<Bc, 32, 0, stream>>>(ps1, S1);
  k3_shift<<<Bc * (Nc / 256), 256, 0, stream>>>(p, S1, w_prev, g, s, gamma, wp, ps2);
  k4_red2<<<Bc, 32, 0, stream>>>(ps2, S2);
  k5_rw<<<Bc * (Nc / 256), 256, 0, stream>>>(memory, wp, S2, e, a, out_mem, rp);
  k6_redr<<<Bc, Mc, 0, stream>>>(rp, out_r);
}